// AttnFusion_7370163880512
// MI455X (gfx1250) — compile-verified
//
#include <hip/hip_runtime.h>

#define XY (512 * 512)   // 262144 spatial cells, power of two (shift 18)
#define DD 128           // feature dim
#define LDSW 132         // padded LDS row stride (132 mod 64 = 4 -> conflict-free)

typedef float v2f __attribute__((ext_vector_type(2)));
typedef float v8f __attribute__((ext_vector_type(8)));

// ---------------- scatter bookkeeping ----------------

__global__ void init_slot(int* slot, int n) {
    int i = blockIdx.x * blockDim.x + threadIdx.x;
    int stride = gridDim.x * blockDim.x;
    for (; i < n; i += stride) slot[i] = -1;
}

__global__ void fill_slot(const int* __restrict__ flat, int* __restrict__ slot, int N) {
    int i = blockIdx.x * blockDim.x + threadIdx.x;
    if (i >= N) return;
    int f = flat[i];
    int cav = f >> 18;          // f / XY
    int s = f & (XY - 1);       // f % XY
    slot[(s << 2) | cav] = i;   // flat_idx unique -> no write conflicts
}

// exclusive scan of occupancy; 256 blocks x 256 threads x 4 elems = XY
__global__ void scan1(const int* __restrict__ slot, int* __restrict__ rank,
                      int* __restrict__ bsum) {
    __shared__ int sh[256];
    int t = threadIdx.x;
    int base = blockIdx.x * 1024 + t * 4;
    int o[4];
    int tsum = 0;
#pragma unroll
    for (int i = 0; i < 4; i++) {
        const int* sp = slot + ((base + i) << 2);
        int occ = (sp[0] >= 0) | (sp[1] >= 0) | (sp[2] >= 0) | (sp[3] >= 0);
        o[i] = occ;
        tsum += occ;
    }
    sh[t] = tsum;
    __syncthreads();
    for (int off = 1; off < 256; off <<= 1) {
        int v = (t >= off) ? sh[t - off] : 0;
        __syncthreads();
        sh[t] += v;
        __syncthreads();
    }
    int run = sh[t] - tsum;  // exclusive within block
#pragma unroll
    for (int i = 0; i < 4; i++) { rank[base + i] = run; run += o[i]; }
    if (t == 255) bsum[blockIdx.x] = sh[255];
}

__global__ void scan2(int* bsum, int* nuniq) {
    __shared__ int sh[256];
    int t = threadIdx.x;
    int v0 = bsum[t];
    sh[t] = v0;
    __syncthreads();
    for (int off = 1; off < 256; off <<= 1) {
        int v = (t >= off) ? sh[t - off] : 0;
        __syncthreads();
        sh[t] += v;
        __syncthreads();
    }
    bsum[t] = sh[t] - v0;           // exclusive block offsets
    if (t == 255) *nuniq = sh[255]; // total unique count
}

__global__ void build_uniq(const int* __restrict__ slot, const int* __restrict__ rank,
                           const int* __restrict__ bsum, int* __restrict__ uniq) {
    int s = blockIdx.x * blockDim.x + threadIdx.x;  // grid == XY exactly
    const int* sp = slot + (s << 2);
    if ((sp[0] >= 0) | (sp[1] >= 0) | (sp[2] >= 0) | (sp[3] >= 0))
        uniq[rank[s] + bsum[s >> 10]] = s;
}

// ---------------- 4-cell attention per wave via f32 WMMA Gram ----------------

__global__ void __launch_bounds__(32)
attn_wmma(const float* __restrict__ feat, const int* __restrict__ slot,
          const int* __restrict__ uniq, const int* __restrict__ nuniq_p,
          float* __restrict__ out, int N) {
    __shared__ float XX[16 * LDSW];  // 16 rows (4 cells x 4 cavs) x 128 feats
    __shared__ float GR[16 * 16];    // Gram spill
    __shared__ float W[16];          // softmax weights
    int lane = threadIdx.x;
    int j0 = blockIdx.x * 4;
    int nun = *nuniq_p;

    // per-row point indices (uniform across lanes)
    int p[16];
#pragma unroll
    for (int r = 0; r < 16; r++) {
        int j = j0 + (r >> 2);
        int s = (j < N && j < nun) ? uniq[j] : (XY - 1);  // padding rows -> ctx[XY-1]
        p[r] = slot[(s << 2) | (r & 3)];
    }
    // stage tile: zeros for absent (cav,cell) rows
#pragma unroll
    for (int r = 0; r < 16; r++) {
        bool ok = p[r] >= 0;
        const float* src = feat + (long)(ok ? p[r] : 0) * DD;  // always in-bounds
#pragma unroll
        for (int c = 0; c < 4; c++) {
            int d = lane + c * 32;
            XX[r * LDSW + d] = ok ? src[d] : 0.0f;
        }
    }
    __syncthreads();

    // Gram = XX * XX^T via V_WMMA_F32_16X16X4_F32 over K=128.
    // For the symmetric Gram, A and B VGPR contents coincide:
    //   A: lane L holds A[M=L%16, K={0,1}|{2,3}]   (halves split at lane 16)
    //   B: lane L holds B[K={0,1}|{2,3}, N=L%16]  == XX[L%16, k+..] too
    v8f acc = {0.f, 0.f, 0.f, 0.f, 0.f, 0.f, 0.f, 0.f};
    int m = lane & 15;
    const float* row = &XX[m * LDSW + ((lane < 16) ? 0 : 2)];
#pragma unroll 4
    for (int k = 0; k < DD; k += 4) {
        v2f av;
        av.x = row[k];
        av.y = row[k + 1];
        acc = __builtin_amdgcn_wmma_f32_16x16x4_f32(false, av, false, av,
                                                    (short)0, acc, false, false);
    }
    // spill D: VGPR j -> rows M=j (lanes 0-15) / M=j+8 (lanes 16-31), N=lane%16
    {
        union { v8f v; float f[8]; } u;
        u.v = acc;
        int rbase = (lane < 16) ? 0 : 8;
#pragma unroll
        for (int jj = 0; jj < 8; jj++) GR[(rbase + jj) * 16 + m] = u.f[jj];
    }
    __syncthreads();

    // softmax over the 4 scores of each cell (query = cav0 row of each 4x4 block)
    if (lane < 16) {
        int li = lane >> 2, c = lane & 3;
        const float scale = 0.0883883476483184405f;  // 1/sqrt(128)
        int rq = li << 2;
        float s0 = GR[rq * 16 + rq + 0] * scale;
        float s1 = GR[rq * 16 + rq + 1] * scale;
        float s2 = GR[rq * 16 + rq + 2] * scale;
        float s3 = GR[rq * 16 + rq + 3] * scale;
        float mx = fmaxf(fmaxf(s0, s1), fmaxf(s2, s3));
        float e0 = expf(s0 - mx), e1 = expf(s1 - mx);
        float e2 = expf(s2 - mx), e3 = expf(s3 - mx);
        float inv = 1.0f / (e0 + e1 + e2 + e3);
        float ec = (c == 0) ? e0 : (c == 1) ? e1 : (c == 2) ? e2 : e3;
        W[lane] = ec * inv;
    }
    __syncthreads();

    // ctx[cell] = sum_c W[c] * XX[cell,c,:], coalesced store
#pragma unroll
    for (int li = 0; li < 4; li++) {
        int j = j0 + li;
        if (j >= N) continue;
        float w0 = W[li * 4 + 0], w1 = W[li * 4 + 1];
        float w2 = W[li * 4 + 2], w3 = W[li * 4 + 3];
#pragma unroll
        for (int c = 0; c < 4; c++) {
            int d = lane + c * 32;
            float v = w0 * XX[(li * 4 + 0) * LDSW + d] + w1 * XX[(li * 4 + 1) * LDSW + d] +
                      w2 * XX[(li * 4 + 2) * LDSW + d] + w3 * XX[(li * 4 + 3) * LDSW + d];
            out[(long)j * DD + d] = v;
        }
    }
}

// ---------------- deterministic segment-mean of coords (<=4 members via slot) ----

__global__ void mean_coor_k(const float* __restrict__ coor, const int* __restrict__ slot,
                            const int* __restrict__ uniq, const int* __restrict__ nuniq_p,
                            float* __restrict__ out, int N) {
    int j = blockIdx.x * blockDim.x + threadIdx.x;
    if (j >= N) return;
    int nun = *nuniq_p;
    float sy = 0.f, sz = 0.f, cnt = 0.f;
    if (j < nun) {
        int s = uniq[j];
#pragma unroll
        for (int c = 0; c < 4; c++) {  // slot order == ascending point index
            int p = slot[(s << 2) | c];
            if (p >= 0) {
                sy += coor[p * 3 + 1];
                sz += coor[p * 3 + 2];
                cnt += 1.0f;
            }
        }
    }
    float inv = 1.0f / fmaxf(cnt, 1.0f);
    float* o = out + (long)j * 3;
    o[0] = 0.0f;      // coor[:,0] zeroed in reference
    o[1] = sy * inv;
    o[2] = sz * inv;  // empty segments -> 0/1 = 0, matches reference
}

// ---------------- launcher ----------------

extern "C" void kernel_launch(void* const* d_in, const int* in_sizes, int n_in,
                              void* d_out, int out_size, void* d_ws, size_t ws_size,
                              hipStream_t stream) {
    const float* feat = (const float*)d_in[0];
    const float* coor = (const float*)d_in[1];
    const int* flat = (const int*)d_in[2];
    int N = in_sizes[0] / DD;

    int* ws = (int*)d_ws;
    int* slot  = ws;                   // 4*XY ints (point index per (spatial,cav))
    int* rank  = ws + 4 * XY;          // XY ints  (in-block exclusive occupancy scan)
    int* bsum  = ws + 5 * XY;          // 256 ints (padded to 512)
    int* uniq  = ws + 5 * XY + 512;    // N ints   (sorted unique spatial values)
    int* nuniq = uniq + N;             // 1 int
    float* out = (float*)d_out;

    hipLaunchKernelGGL(init_slot, dim3(1024), dim3(256), 0, stream, slot, 4 * XY);
    hipLaunchKernelGGL(fill_slot, dim3((N + 255) / 256), dim3(256), 0, stream, flat, slot, N);
    hipLaunchKernelGGL(scan1, dim3(256), dim3(256), 0, stream, slot, rank, bsum);
    hipLaunchKernelGGL(scan2, dim3(1), dim3(256), 0, stream, bsum, nuniq);
    hipLaunchKernelGGL(build_uniq, dim3(XY / 256), dim3(256), 0, stream, slot, rank, bsum, uniq);
    hipLaunchKernelGGL(attn_wmma, dim3((N + 3) / 4), dim3(32), 0, stream,
                       feat, slot, uniq, nuniq, out, N);
    hipLaunchKernelGGL(mean_coor_k, dim3((N + 255) / 256), dim3(256), 0, stream,
                       coor, slot, uniq, nuniq, out + (size_t)N * DD, N);
}